// GCN_39591008534906
// MI455X (gfx1250) — compile-verified
//
#include <hip/hip_runtime.h>
#include <hip/hip_bf16.h>
#include <math.h>

typedef __attribute__((ext_vector_type(2))) float v2f;
typedef __attribute__((ext_vector_type(8))) float v8f;

// ---------------------------------------------------------------------------
// Degree / normalization kernels
// ---------------------------------------------------------------------------
__global__ void k_deg_init(float* __restrict__ deg, int N) {
    int i = blockIdx.x * blockDim.x + threadIdx.x;
    if (i < N) deg[i] = 1.0f;  // self-loop contributes 1 to every node's degree
}

__global__ void k_deg_count(const int* __restrict__ ei, float* __restrict__ deg,
                            long long E) {
    long long e = (long long)blockIdx.x * blockDim.x + threadIdx.x;
    if (e < E) {
        int d = ei[E + e];  // edge_index[1][e]
        atomicAdd(&deg[d], 1.0f);
    }
}

__global__ void k_rsqrt_inplace(float* __restrict__ deg, int N) {
    int i = blockIdx.x * blockDim.x + threadIdx.x;
    if (i < N) deg[i] = rsqrtf(deg[i]);  // deg >= 1 always (self-loops)
}

// ---------------------------------------------------------------------------
// WMMA f32 GEMM:  out[N, ldOut] = preop(A[N,128]) @ W[128, Fout]
// preop = relu(a + biasPrev[k]) when RELU (bias of the PREVIOUS layer,
// applied post-aggregation per the reference, fused into the A staging).
//
// Block: 128 threads = 4 waves; each wave computes 16 rows x NT*16 cols.
// A (64x128 block tile) and W (128 x NT*16, zero-padded) staged in LDS,
// PAIR-INTERLEAVED:  s[pair p][n] = {M[2p][n], M[2p+1][n]}  -> each lane's
// WMMA fragment is one aligned ds_load_b64 into an even VGPR pair.
// Inner loop is software-pipelined: next k-step's fragments are fetched
// into a second register set while the current 8 WMMAs issue.
// ---------------------------------------------------------------------------
template <int NT, bool RELU>
__global__ void k_gemm_wmma(const float* __restrict__ A,
                            const float* __restrict__ W,
                            const float* __restrict__ biasPrev,
                            float* __restrict__ out,
                            int N, int Fout, int ldOut) {
    constexpr int LDW = NT * 16;          // B columns held in LDS
    extern __shared__ char smem[];
    v2f* sB = (v2f*)smem;                           // [64 pairs][LDW]
    v2f* sA = (v2f*)(smem + (size_t)64 * LDW * 8);  // [4 waves][64 pairs][16 rows]

    const int tid  = threadIdx.x;
    const int rowB = blockIdx.x * 64;

    // ---- stage B = W (zero-padded), pair-interleaved ----
    for (int idx = tid; idx < 64 * LDW; idx += blockDim.x) {
        const int p = idx / LDW, c = idx % LDW;
        v2f w = {};
        if (c < Fout) {
            w.x = W[(2 * p) * Fout + c];
            w.y = W[(2 * p + 1) * Fout + c];
        }
        sB[idx] = w;
    }

    // ---- stage A block tile (64 rows x 64 pairs), coalesced b64 loads ----
    for (int idx = tid; idx < 64 * 64; idx += blockDim.x) {
        const int r   = idx >> 6;   // row within block tile
        const int p   = idx & 63;   // K pair
        const int row = rowB + r;
        const int rowc = (row < N) ? row : (N - 1);
        v2f a = *(const v2f*)(A + (size_t)rowc * 128 + 2 * p);
        if (RELU) {
            a.x = fmaxf(a.x + biasPrev[2 * p], 0.0f);
            a.y = fmaxf(a.y + biasPrev[2 * p + 1], 0.0f);
        }
        if (row >= N) { a.x = 0.0f; a.y = 0.0f; }
        sA[(r >> 4) * 1024 + p * 16 + (r & 15)] = a;
    }
    __syncthreads();

    const int lane = tid & 31;
    const int wave = tid >> 5;
    const int m    = lane & 15;   // row/col within 16x16 tile
    const int kh   = lane >> 4;   // 0/1 -> K pair offset +0 / +1
    const int waveOff = wave * 1024;

    v8f acc[NT] = {};

    // ---- software-pipelined K loop: 32 steps of K=4 ----
    v2f aCur, bCur[NT];
    {
        const int p = kh;
        aCur = sA[waveOff + p * 16 + m];
#pragma unroll
        for (int t = 0; t < NT; ++t) bCur[t] = sB[p * LDW + t * 16 + m];
    }

#pragma unroll 8
    for (int k2 = 0; k2 < 62; k2 += 2) {
        // prefetch next step's fragments while current WMMAs issue
        const int pn = k2 + 2 + kh;
        v2f aNxt = sA[waveOff + pn * 16 + m];
        v2f bNxt[NT];
#pragma unroll
        for (int t = 0; t < NT; ++t) bNxt[t] = sB[pn * LDW + t * 16 + m];

#pragma unroll
        for (int t = 0; t < NT; ++t) {
            acc[t] = __builtin_amdgcn_wmma_f32_16x16x4_f32(
                false, aCur, false, bCur[t], (short)0, acc[t], false, false);
        }
        aCur = aNxt;
#pragma unroll
        for (int t = 0; t < NT; ++t) bCur[t] = bNxt[t];
    }
    // epilogue step
#pragma unroll
    for (int t = 0; t < NT; ++t) {
        acc[t] = __builtin_amdgcn_wmma_f32_16x16x4_f32(
            false, aCur, false, bCur[t], (short)0, acc[t], false, false);
    }

    // C layout: VGPR r holds (M=r, N=lane) for lanes 0-15, (M=r+8) for 16-31.
    const int row0 = rowB + wave * 16;
#pragma unroll
    for (int t = 0; t < NT; ++t) {
        const int col = t * 16 + m;
#pragma unroll
        for (int r = 0; r < 8; ++r) {
            const int orow = row0 + r + 8 * kh;
            if (orow < N) out[(size_t)orow * ldOut + col] = acc[t][r];
        }
    }
}

// ---------------------------------------------------------------------------
// Aggregation: self-loop term INITIALIZES the output (out = t * dis^2),
// then the edge kernel atomically accumulates messages on top.
// ---------------------------------------------------------------------------
__global__ void k_selfloop_init(const float* __restrict__ t,
                                const float* __restrict__ dis,
                                float* __restrict__ out,
                                long long total, int F) {
    long long i = (long long)blockIdx.x * blockDim.x + threadIdx.x;
    if (i < total) {
        int node = (int)(i / F);
        float s = dis[node];
        out[i] = t[i] * s * s;
    }
}

template <int F>
__global__ void k_scatter_edges(const int* __restrict__ ei,
                                const float* __restrict__ t,
                                const float* __restrict__ dis,
                                float* __restrict__ out, long long E) {
    constexpr int EP = 256 / F;  // edges per 256-thread block
    long long e = (long long)blockIdx.x * EP + threadIdx.x / F;
    int f = threadIdx.x % F;
    if (e < E) {
        int s = ei[e];
        int d = ei[E + e];
        float nrm = dis[s] * dis[d];
        atomicAdd(&out[(size_t)d * F + f], t[(size_t)s * F + f] * nrm);
    }
}

// ---------------------------------------------------------------------------
// Final: out[i, c] = log_softmax(a2[i, 0:10] + bout)  (a2 has stride 16)
// ---------------------------------------------------------------------------
__global__ void k_logsoftmax(const float* __restrict__ a2,
                             const float* __restrict__ bout,
                             float* __restrict__ out, int N) {
    int i = blockIdx.x * blockDim.x + threadIdx.x;
    if (i >= N) return;
    float v[10];
    float mx = -INFINITY;
#pragma unroll
    for (int c = 0; c < 10; ++c) {
        v[c] = a2[(size_t)i * 16 + c] + bout[c];
        mx = fmaxf(mx, v[c]);
    }
    float sum = 0.0f;
#pragma unroll
    for (int c = 0; c < 10; ++c) sum += expf(v[c] - mx);
    float lse = logf(sum);
#pragma unroll
    for (int c = 0; c < 10; ++c) out[(size_t)i * 10 + c] = v[c] - mx - lse;
}

// ---------------------------------------------------------------------------
// Host orchestration
// ---------------------------------------------------------------------------
extern "C" void kernel_launch(void* const* d_in, const int* in_sizes, int n_in,
                              void* d_out, int out_size, void* d_ws, size_t ws_size,
                              hipStream_t stream) {
    const float* x    = (const float*)d_in[0];
    const int*   ei   = (const int*)d_in[1];   // int32 (JAX x64 disabled)
    const float* W0   = (const float*)d_in[2];
    const float* b0   = (const float*)d_in[3];
    const float* W1   = (const float*)d_in[4];
    const float* b1   = (const float*)d_in[5];
    const float* Wout = (const float*)d_in[6];
    const float* bout = (const float*)d_in[7];

    const int N = in_sizes[0] / 128;
    const long long E = in_sizes[1] / 2;

    auto al = [](size_t v) { return (v + 255) & ~(size_t)255; };
    char* ws = (char*)d_ws;
    size_t off = 0;
    float* dis = (float*)(ws + off); off += al((size_t)N * 4);
    float* B1  = (float*)(ws + off); off += al((size_t)N * 128 * 4);
    float* B2  = (float*)(ws + off); off += al((size_t)N * 128 * 4);
    float* B3  = (float*)(ws + off); off += al((size_t)N * 16 * 4);
    float* A2  = B1;  // N x 16, reuses B1 (free by the time it's needed)

    const int T = 256;
    const long long NF  = (long long)N * 128;
    const long long N16 = (long long)N * 16;

    // dynamic LDS:  B pairs (64*NT*16*8) + A pairs (4*64*16*8 = 32768)
    const size_t smem8 = (size_t)64 * 128 * 8 + 32768;  // 96 KB (NT=8)
    const size_t smem1 = (size_t)64 * 16 * 8 + 32768;   // 40 KB (NT=1)
    const int gemmGrid = (N + 63) / 64;

    // 1) normalization: deg -> dis
    k_deg_init<<<(N + T - 1) / T, T, 0, stream>>>(dis, N);
    k_deg_count<<<(int)((E + T - 1) / T), T, 0, stream>>>(ei, dis, E);
    k_rsqrt_inplace<<<(N + T - 1) / T, T, 0, stream>>>(dis, N);

    // 2) layer 0: t0 = x @ W0 ; a0 = aggregate(t0)        (B1=t0, B2=a0)
    k_gemm_wmma<8, false><<<gemmGrid, 128, smem8, stream>>>(x, W0, nullptr, B1, N, 128, 128);
    k_selfloop_init<<<(int)((NF + T - 1) / T), T, 0, stream>>>(B1, dis, B2, NF, 128);
    k_scatter_edges<128><<<(int)((E + 1) / 2), T, 0, stream>>>(ei, B1, dis, B2, E);

    // 3) layer 1: t1 = relu(a0+b0) @ W1 ; a1 = aggregate  (B1=t1, B2=a1)
    k_gemm_wmma<8, true><<<gemmGrid, 128, smem8, stream>>>(B2, W1, b0, B1, N, 128, 128);
    k_selfloop_init<<<(int)((NF + T - 1) / T), T, 0, stream>>>(B1, dis, B2, NF, 128);
    k_scatter_edges<128><<<(int)((E + 1) / 2), T, 0, stream>>>(ei, B1, dis, B2, E);

    // 4) layer 2: t2 = relu(a1+b1) @ Wout (10 cols, padded to 16)  (B3=t2)
    k_gemm_wmma<1, true><<<gemmGrid, 128, smem1, stream>>>(B2, Wout, b1, B3, N, 10, 16);
    k_selfloop_init<<<(int)((N16 + T - 1) / T), T, 0, stream>>>(B3, dis, A2, N16, 16);
    k_scatter_edges<16><<<(int)((E + 15) / 16), T, 0, stream>>>(ei, B3, dis, A2, E);

    // 5) log_softmax(a2 + bout)
    k_logsoftmax<<<(N + T - 1) / T, T, 0, stream>>>(A2, bout, (float*)d_out, N);
}